// FPNN_v3_84061099917748
// MI455X (gfx1250) — compile-verified
//
#include <hip/hip_runtime.h>
#include <hip/hip_bf16.h>
#include <math.h>
#include <stdint.h>

typedef __attribute__((ext_vector_type(2))) float v2f;
typedef __attribute__((ext_vector_type(8))) float v8f;

#define IN_CH   128
#define OUT_CH  64
#define MAX_DEG 4
#define NBUCKET 5
#define KP      32      // K panel size
#define PADA    36      // padded A stride (floats): 16 rows hit 16 distinct bank pairs
#define WROW    160     // swizzled W k-pair row: 128 data floats + 32 pad -> +32 bank shift

// ---------------------------------------------------------------------------
// Kernel 1: edge scatter-add.  One wave32 per edge; lane l handles channels
// [4l..4l+3] via a single 16B load + 4 global f32 atomics. Lane 0 bumps degree.
// ---------------------------------------------------------------------------
__global__ __launch_bounds__(256)
void mf_edge_scatter(const float* __restrict__ x,
                     const int* __restrict__ ei,
                     float* __restrict__ agg,
                     int* __restrict__ deg,
                     int nE) {
    int t = blockIdx.x * blockDim.x + threadIdx.x;
    int e = t >> 5;
    int lane = t & 31;
    if (e >= nE) return;
    int src = ei[e];
    int dst = ei[nE + e];
    const float4* xs = (const float4*)(x + (size_t)src * IN_CH);
    float4 v = xs[lane];
    float* ag = agg + (size_t)dst * IN_CH + lane * 4;
    atomicAdd(ag + 0, v.x);
    atomicAdd(ag + 1, v.y);
    atomicAdd(ag + 2, v.z);
    atomicAdd(ag + 3, v.w);
    if (lane == 0) atomicAdd(deg + dst, 1);
}

// ---------------------------------------------------------------------------
// Kernel 2: compact nodes into 5 degree buckets (deg clamped to 4).
// ---------------------------------------------------------------------------
__global__ __launch_bounds__(256)
void mf_bucket(const int* __restrict__ deg,
               int* __restrict__ counts,
               int* __restrict__ lists,
               int n) {
    int i = blockIdx.x * blockDim.x + threadIdx.x;
    if (i >= n) return;
    int d = deg[i];
    if (d > MAX_DEG) d = MAX_DEG;
    int pos = atomicAdd(counts + d, 1);
    lists[(size_t)d * n + pos] = i;
}

// ---------------------------------------------------------------------------
// Kernel 3: bucketed WMMA GEMM + relu + row softmax + pooled atomic scatter.
// 4 waves per WG; wave w owns a 16-node tile from bucket d (uniform weights).
// out[16,64] = agg[16,128] @ W_lin[d] + x[16,128] @ W_root[d]
// V_WMMA_F32_16X16X4_F32; K panels of 32 staged in LDS.
//   * W panel stored pair-interleaved so each B fragment is one ds_load_b64.
//   * A tiles staged with GLOBAL_LOAD_ASYNC_TO_LDS_B128 (ASYNCcnt-tracked).
// ---------------------------------------------------------------------------
__global__ __launch_bounds__(128)
void mf_gemm(const float* __restrict__ x,
             const float* __restrict__ agg,
             const int* __restrict__ batch,
             const float* __restrict__ Wlin,
             const float* __restrict__ Wroot,
             const int* __restrict__ counts,
             const int* __restrict__ lists,
             float* __restrict__ pooled,
             int n, int tileGroups) {
    __shared__ float sWp[2][KP / 2][WROW];   // [mat][k>>1][n*2 + (k&1)]
    __shared__ float sA[4][2][16][PADA];     // [wave][mat][m][k]
    __shared__ int   sNode[4][16];
    __shared__ int   sBatch[4][16];

    int d = blockIdx.x / tileGroups;         // bucket (block-uniform)
    int g = blockIdx.x % tileGroups;
    int cnt = counts[d];
    if (g * 64 >= cnt) return;               // block-uniform early exit

    int tid  = threadIdx.x;
    int w    = tid >> 5;                     // wave id 0..3
    int lane = tid & 31;
    int half = lane >> 4;                    // 0 | 1
    int lm   = lane & 15;

    int rowBase = (g * 4 + w) * 16;

    if (lane < 16) {
        int idx = rowBase + lane;
        int node = (idx < cnt) ? lists[(size_t)d * n + idx] : -1;
        sNode[w][lane]  = node;
        sBatch[w][lane] = (node >= 0) ? batch[node] : 0;
    }
    __syncthreads();

    const float* WlinD  = Wlin  + (size_t)d * IN_CH * OUT_CH;
    const float* WrootD = Wroot + (size_t)d * IN_CH * OUT_CH;

    // per-lane A staging geometry: 2 lanes per node row, 16 floats per lane
    int arow  = lane >> 1;
    int acolb = (lane & 1) * 16;
    int anode = sNode[w][arow];
    int ann   = anode < 0 ? 0 : anode;
    unsigned aByte = (unsigned)ann * (IN_CH * 4);

    v8f c0 = {}, c1 = {}, c2 = {}, c3 = {};

    for (int kp = 0; kp < IN_CH; kp += KP) {
        __syncthreads();
        // ---- cooperative W panel load -> pair-interleaved LDS ----
        #pragma unroll
        for (int j = 0; j < (KP * OUT_CH) / 128; ++j) {
            int idx = tid + j * 128;         // 0..2047
            int r   = idx >> 6;              // k within panel
            int cix = idx & 63;              // n
            float vl = WlinD[(kp + r) * OUT_CH + cix];
            float vr = WrootD[(kp + r) * OUT_CH + cix];
            sWp[0][r >> 1][cix * 2 + (r & 1)] = vl;
            sWp[1][r >> 1][cix * 2 + (r & 1)] = vr;
        }
        // ---- A tiles: async global->LDS 16B copies, then drain ASYNCcnt ----
        {
            unsigned gOff = aByte + (unsigned)(kp + acolb) * 4;
            #pragma unroll
            for (int j = 0; j < 4; ++j) {
                unsigned l0 = (unsigned)(uintptr_t)&sA[w][0][arow][acolb + 4 * j];
                unsigned l1 = (unsigned)(uintptr_t)&sA[w][1][arow][acolb + 4 * j];
                unsigned go = gOff + 16u * j;
                asm volatile("global_load_async_to_lds_b128 %0, %1, %2 offset:0"
                             :: "v"(l0), "v"(go), "s"(agg) : "memory");
                asm volatile("global_load_async_to_lds_b128 %0, %1, %2 offset:0"
                             :: "v"(l1), "v"(go), "s"(x) : "memory");
            }
            asm volatile("s_wait_asynccnt 0x0" ::: "memory");
            if (anode < 0) {                 // zero padding rows (bucket tail only)
                #pragma unroll
                for (int j = 0; j < 16; ++j) {
                    sA[w][0][arow][acolb + j] = 0.f;
                    sA[w][1][arow][acolb + j] = 0.f;
                }
            }
        }
        __syncthreads();

        // ---- WMMA over this K panel: A & B fragments are single b64 LDS reads ----
        #pragma unroll
        for (int kk = 0; kk < KP; kk += 4) {
            int k2 = (kk >> 1) + half;       // lane's k-pair row
            #pragma unroll
            for (int mat = 0; mat < 2; ++mat) {
                v2f a = *(const v2f*)&sA[w][mat][lm][kk + 2 * half];
                const float* bp = &sWp[mat][k2][2 * lm];
                v2f b0 = *(const v2f*)(bp + 0);
                v2f b1 = *(const v2f*)(bp + 32);
                v2f b2 = *(const v2f*)(bp + 64);
                v2f b3 = *(const v2f*)(bp + 96);
                c0 = __builtin_amdgcn_wmma_f32_16x16x4_f32(false, a, false, b0,
                                                           (short)0, c0, false, false);
                c1 = __builtin_amdgcn_wmma_f32_16x16x4_f32(false, a, false, b1,
                                                           (short)0, c1, false, false);
                c2 = __builtin_amdgcn_wmma_f32_16x16x4_f32(false, a, false, b2,
                                                           (short)0, c2, false, false);
                c3 = __builtin_amdgcn_wmma_f32_16x16x4_f32(false, a, false, b3,
                                                           (short)0, c3, false, false);
            }
        }
    }

    // ---- epilogue: relu -> row softmax (64 cols) -> pooled atomic add ----
    // C/D layout: VGPR r, lane l holds element (M = r + 8*half, N = 16*nb + lm).
    #pragma unroll
    for (int r = 0; r < 8; ++r) {
        int m = r + 8 * half;
        float v0 = fmaxf(c0[r], 0.f);
        float v1 = fmaxf(c1[r], 0.f);
        float v2 = fmaxf(c2[r], 0.f);
        float v3 = fmaxf(c3[r], 0.f);
        float mx = fmaxf(fmaxf(v0, v1), fmaxf(v2, v3));
        #pragma unroll
        for (int s = 1; s < 16; s <<= 1) mx = fmaxf(mx, __shfl_xor(mx, s, 16));
        float e0 = __expf(v0 - mx), e1 = __expf(v1 - mx);
        float e2 = __expf(v2 - mx), e3 = __expf(v3 - mx);
        float sum = e0 + e1 + e2 + e3;
        #pragma unroll
        for (int s = 1; s < 16; s <<= 1) sum += __shfl_xor(sum, s, 16);
        float inv = 1.f / sum;
        int node = sNode[w][m];
        if (node >= 0) {
            float* pr = pooled + (size_t)sBatch[w][m] * OUT_CH;
            atomicAdd(pr + 0 * 16 + lm, e0 * inv);
            atomicAdd(pr + 1 * 16 + lm, e1 * inv);
            atomicAdd(pr + 2 * 16 + lm, e2 * inv);
            atomicAdd(pr + 3 * 16 + lm, e3 * inv);
        }
    }
}

// ---------------------------------------------------------------------------
extern "C" void kernel_launch(void* const* d_in, const int* in_sizes, int n_in,
                              void* d_out, int out_size, void* d_ws, size_t ws_size,
                              hipStream_t stream) {
    const float* x     = (const float*)d_in[0];
    const int*   ei    = (const int*)d_in[1];
    const int*   batch = (const int*)d_in[2];
    const float* Wlin  = (const float*)d_in[3];
    const float* Wroot = (const float*)d_in[4];
    float* pooled = (float*)d_out;

    int n  = in_sizes[0] / IN_CH;   // 200000
    int nE = in_sizes[1] / 2;       // 800000

    // workspace layout
    float* agg   = (float*)d_ws;                       // n*128 floats
    int*   deg   = (int*)(agg + (size_t)n * IN_CH);    // n ints
    int*   cnts  = deg + n;                            // 8 ints (5 used)
    int*   lists = cnts + 8;                           // 5*n ints

    hipMemsetAsync(agg, 0, (size_t)n * IN_CH * sizeof(float), stream);
    hipMemsetAsync(deg, 0, ((size_t)n + 8) * sizeof(int), stream);
    hipMemsetAsync(d_out, 0, (size_t)out_size * sizeof(float), stream);

    {
        long long threads = (long long)nE * 32;
        int blocks = (int)((threads + 255) / 256);
        mf_edge_scatter<<<blocks, 256, 0, stream>>>(x, ei, agg, deg, nE);
    }
    mf_bucket<<<(n + 255) / 256, 256, 0, stream>>>(deg, cnts, lists, n);

    int tileGroups = (n + 63) / 64;
    mf_gemm<<<NBUCKET * tileGroups, 128, 0, stream>>>(
        x, agg, batch, Wlin, Wroot, cnts, lists, pooled, n, tileGroups);
}